// PointNetSetAbstraction_89721866814223
// MI455X (gfx1250) — compile-verified
//
#include <hip/hip_runtime.h>

#define NPT    1024
#define NSAMP  32
#define NB     8
#define NN     16384
#define RAD2   0.04f
#define EPS_BN 1e-5f

typedef __attribute__((ext_vector_type(2))) float v2f;
typedef __attribute__((ext_vector_type(8))) float v8f;

// ---------------- BN/bias folding: A = g*rsqrt(var+eps), B = A*(bias-mean)+beta
__global__ void coef_kernel(const float* __restrict__ bias, const float* __restrict__ gamma,
                            const float* __restrict__ beta, const float* __restrict__ mean,
                            const float* __restrict__ var, float* __restrict__ A,
                            float* __restrict__ Bc, int n) {
  int i = blockIdx.x * blockDim.x + threadIdx.x;
  if (i < n) {
    float s = gamma[i] * rsqrtf(var[i] + EPS_BN);
    A[i]  = s;
    Bc[i] = s * (bias[i] - mean[i]) + beta[i];
  }
}

// ---------------- Farthest point sampling: one workgroup per batch, all-LDS
__global__ __launch_bounds__(1024)
void fps_kernel(const float* __restrict__ xyz, float* __restrict__ newxyz_ws,
                float* __restrict__ out_newxyz) {
  extern __shared__ float smem[];
  float* xs   = smem;            // NN
  float* ys   = xs + NN;         // NN
  float* zs   = ys + NN;         // NN
  float* dist = zs + NN;         // NN
  float* rval = dist + NN;       // 32
  int*   ridx = (int*)(rval + 32);
  __shared__ int   s_cur;
  __shared__ float s_c[3];

  const int b   = blockIdx.x;
  const int tid = threadIdx.x;
  const float* xb = xyz + (size_t)b * 3 * NN;

  for (int p = tid; p < NN; p += 1024) {
    xs[p] = xb[p];
    ys[p] = xb[NN + p];
    zs[p] = xb[2 * NN + p];
    dist[p] = 1e10f;
  }
  if (tid == 0) s_cur = 0;
  __syncthreads();

  const int lane = tid & 31;
  const int wid  = tid >> 5;

  for (int t = 0; t < NPT; ++t) {
    if (tid == 0) {
      int cur = s_cur;
      float cx = xs[cur], cy = ys[cur], cz = zs[cur];
      s_c[0] = cx; s_c[1] = cy; s_c[2] = cz;
      newxyz_ws[((size_t)b * NPT + t) * 3 + 0] = cx;
      newxyz_ws[((size_t)b * NPT + t) * 3 + 1] = cy;
      newxyz_ws[((size_t)b * NPT + t) * 3 + 2] = cz;
      out_newxyz[(size_t)b * 3 * NPT + 0 * NPT + t] = cx;
      out_newxyz[(size_t)b * 3 * NPT + 1 * NPT + t] = cy;
      out_newxyz[(size_t)b * 3 * NPT + 2 * NPT + t] = cz;
    }
    __syncthreads();
    const float cx = s_c[0], cy = s_c[1], cz = s_c[2];
    float bv = -1.0f; int bi = 0x7fffffff;
#pragma unroll
    for (int j = 0; j < NN / 1024; ++j) {
      int p = j * 1024 + tid;
      float dx = xs[p] - cx, dy = ys[p] - cy, dz = zs[p] - cz;
      float d  = dx * dx + dy * dy + dz * dz;
      float dm = fminf(dist[p], d);
      dist[p] = dm;
      if (dm > bv) { bv = dm; bi = p; }   // ascending p -> keeps lowest index on tie
    }
    // wave32 argmax reduction (first-index tie-break to match jnp.argmax)
    for (int off = 16; off > 0; off >>= 1) {
      float ov = __shfl_down(bv, off, 32);
      int   oi = __shfl_down(bi, off, 32);
      if (ov > bv || (ov == bv && oi < bi)) { bv = ov; bi = oi; }
    }
    if (lane == 0) { rval[wid] = bv; ridx[wid] = bi; }
    __syncthreads();
    if (wid == 0) {
      float v = rval[lane]; int i = ridx[lane];
      for (int off = 16; off > 0; off >>= 1) {
        float ov = __shfl_down(v, off, 32);
        int   oi = __shfl_down(i, off, 32);
        if (ov > v || (ov == v && oi < i)) { v = ov; i = oi; }
      }
      if (lane == 0) s_cur = i;
    }
    __syncthreads();
  }
}

// ---------------- Ball query: one wave per centroid; ballot compaction keeps
// ascending index order == sort()[:nsample]; tail filled with first index.
__global__ __launch_bounds__(256)
void ballquery_kernel(const float* __restrict__ xyz, const float* __restrict__ newxyz_ws,
                      int* __restrict__ idx_ws) {
  __shared__ int buf[8][NSAMP];
  const int wid  = threadIdx.x >> 5;
  const int lane = threadIdx.x & 31;
  const int g = blockIdx.x * 8 + wid;      // centroid id
  const int b = g >> 10;
  const float* xb = xyz + (size_t)b * 3 * NN;
  const float cx = newxyz_ws[(size_t)g * 3 + 0];
  const float cy = newxyz_ws[(size_t)g * 3 + 1];
  const float cz = newxyz_ws[(size_t)g * 3 + 2];
  int cnt = 0;
  for (int c = 0; c < NN / 32; ++c) {
    if (cnt >= NSAMP) break;               // wave-uniform early exit
    int p = c * 32 + lane;
    float dx = xb[p] - cx, dy = xb[NN + p] - cy, dz = xb[2 * NN + p] - cz;
    float d2 = dx * dx + dy * dy + dz * dz;
    bool in = (d2 <= RAD2);
    unsigned m = (unsigned)__ballot(in);
    if (in) {
      int pos = cnt + __popc(m & ((1u << lane) - 1u));
      if (pos < NSAMP) buf[wid][pos] = p;
    }
    cnt += __popc(m);
  }
  if (cnt > NSAMP) cnt = NSAMP;
  int first = buf[wid][0];                 // centroid itself always in ball -> cnt>=1
  int v = (lane < cnt) ? buf[wid][lane] : first;
  idx_ws[(size_t)g * NSAMP + lane] = v;
}

// ---------------- WMMA fragment loaders (fp32 16x16x4 layouts per ISA 7.12.2)
// A 16x4: lanes 0-15 M=0..15 with K0,K1 in v0,v1; lanes 16-31 same M with K2,K3.
__device__ __forceinline__ v2f frag_a(const float* h, int stride, int mt, int kt, int lane) {
  const int half = lane >> 4;
  const int row  = mt * 16 + (lane & 15);
  const int k    = kt * 4 + 2 * half;
  v2f a;
  a.x = h[row * stride + k];
  a.y = h[row * stride + k + 1];
  return a;
}
// B 4x16: v0 holds rows {K0 | K2}, v1 holds rows {K1 | K3}, cols across lanes.
__device__ __forceinline__ v2f frag_b(const float* w, int wstride, int nt, int kt, int lane) {
  const int half = lane >> 4;
  const int col  = nt * 16 + (lane & 15);
  const int k    = kt * 4 + 2 * half;
  v2f bb;
  bb.x = w[k * wstride + col];
  bb.y = w[(k + 1) * wstride + col];
  return bb;
}

#define HS 66   // padded LDS row stride (even for b64, breaks 64-bank alignment)

// ---------------- Fused gather + 3-layer MLP (fp32 WMMA) + max-pool over K
__global__ __launch_bounds__(128)
void mlp_kernel(const float* __restrict__ xyz, const float* __restrict__ points,
                const float* __restrict__ W1, const float* __restrict__ W2,
                const float* __restrict__ W3,
                const float* __restrict__ newxyz_ws, const int* __restrict__ idx_ws,
                const float* __restrict__ coefA, const float* __restrict__ coefB,
                float* __restrict__ out_points) {
  extern __shared__ float smem[];
  float* w1s   = smem;                 // 8 x 64 (rows 6,7 zero-padded)
  float* w2s   = w1s + 8 * 64;         // 64 x 64
  float* w3s   = w2s + 64 * 64;        // 64 x 128
  float* cA    = w3s + 64 * 128;       // 256
  float* cB    = cA + 256;             // 256
  float* hbase = cB + 256;             // per-wave: h0(32x8) h1(32xHS) h2(32xHS)

  const int tid  = threadIdx.x;
  const int lane = tid & 31;
  const int wid  = tid >> 5;

  // cooperative weight / coef staging
  for (int i = tid; i < 8 * 64; i += 128) w1s[i] = ((i >> 6) < 6) ? W1[i] : 0.f;
  for (int i = tid; i < 64 * 64; i += 128) w2s[i] = W2[i];
  for (int i = tid; i < 64 * 128; i += 128) w3s[i] = W3[i];
  for (int i = tid; i < 256; i += 128) { cA[i] = coefA[i]; cB[i] = coefB[i]; }

  const int g = blockIdx.x * 4 + wid;  // s-group (b,s): 32 samples = 2 M-tiles
  const int b = g >> 10;
  const int s = g & 1023;

  float* h0 = hbase + wid * (256 + 2 * 32 * HS);
  float* h1 = h0 + 256;
  float* h2 = h1 + 32 * HS;

  // gather input features: [pts(3) | xyz - centroid(3) | 0 0]
  {
    const float* pb = points + (size_t)b * 3 * NN;
    const float* xb = xyz + (size_t)b * 3 * NN;
    const float cx = newxyz_ws[(size_t)g * 3 + 0];
    const float cy = newxyz_ws[(size_t)g * 3 + 1];
    const float cz = newxyz_ws[(size_t)g * 3 + 2];
    const int ip = idx_ws[(size_t)g * NSAMP + lane];
    float* r = h0 + lane * 8;
    r[0] = pb[ip];
    r[1] = pb[NN + ip];
    r[2] = pb[2 * NN + ip];
    r[3] = xb[ip] - cx;
    r[4] = xb[NN + ip] - cy;
    r[5] = xb[2 * NN + ip] - cz;
    r[6] = 0.f;
    r[7] = 0.f;
  }
  __syncthreads();

  // Layer 1: (32x8)x(8x64) -> h1, fused bias+BN+ReLU
  for (int mt = 0; mt < 2; ++mt)
    for (int nt = 0; nt < 4; ++nt) {
      v8f acc = {};
#pragma unroll
      for (int kt = 0; kt < 2; ++kt) {
        v2f a = frag_a(h0, 8, mt, kt, lane);
        v2f w = frag_b(w1s, 64, nt, kt, lane);
        acc = __builtin_amdgcn_wmma_f32_16x16x4_f32(false, a, false, w, (short)0, acc,
                                                    false, false);
      }
      const int half = lane >> 4;
      const int col  = nt * 16 + (lane & 15);
      const float A = cA[col], Bv = cB[col];
#pragma unroll
      for (int i = 0; i < 8; ++i) {
        int row = mt * 16 + i + 8 * half;   // C layout: vgpr i -> M=i / M=i+8
        h1[row * HS + col] = fmaxf(A * acc[i] + Bv, 0.f);
      }
    }

  // Layer 2: (32x64)x(64x64) -> h2
  for (int mt = 0; mt < 2; ++mt)
    for (int nt = 0; nt < 4; ++nt) {
      v8f acc = {};
#pragma unroll
      for (int kt = 0; kt < 16; ++kt) {
        v2f a = frag_a(h1, HS, mt, kt, lane);
        v2f w = frag_b(w2s, 64, nt, kt, lane);
        acc = __builtin_amdgcn_wmma_f32_16x16x4_f32(false, a, false, w, (short)0, acc,
                                                    false, false);
      }
      const int half = lane >> 4;
      const int col  = nt * 16 + (lane & 15);
      const float A = cA[64 + col], Bv = cB[64 + col];
#pragma unroll
      for (int i = 0; i < 8; ++i) {
        int row = mt * 16 + i + 8 * half;
        h2[row * HS + col] = fmaxf(A * acc[i] + Bv, 0.f);
      }
    }

  // Layer 3: (32x64)x(64x128), max over the 32 samples in-register,
  // store transposed (B,128,S). ReLU output >= 0 so 0 is a valid max identity.
  for (int nt = 0; nt < 8; ++nt) {
    const int col = nt * 16 + (lane & 15);
    const float A = cA[128 + col], Bv = cB[128 + col];
    float cmax = 0.f;
    for (int mt = 0; mt < 2; ++mt) {
      v8f acc = {};
#pragma unroll
      for (int kt = 0; kt < 16; ++kt) {
        v2f a = frag_a(h2, HS, mt, kt, lane);
        v2f w = frag_b(w3s, 128, nt, kt, lane);
        acc = __builtin_amdgcn_wmma_f32_16x16x4_f32(false, a, false, w, (short)0, acc,
                                                    false, false);
      }
#pragma unroll
      for (int i = 0; i < 8; ++i)
        cmax = fmaxf(cmax, fmaxf(A * acc[i] + Bv, 0.f));
    }
    cmax = fmaxf(cmax, __shfl_xor(cmax, 16, 32));   // combine M=i and M=i+8 halves
    if (lane < 16)
      out_points[(size_t)b * 128 * NPT + (size_t)col * NPT + s] = cmax;
  }
}

// ---------------- host side
extern "C" void kernel_launch(void* const* d_in, const int* in_sizes, int n_in,
                              void* d_out, int out_size, void* d_ws, size_t ws_size,
                              hipStream_t stream) {
  const float* xyz    = (const float*)d_in[0];
  const float* points = (const float*)d_in[1];
  const float* W1  = (const float*)d_in[2];
  const float* b1  = (const float*)d_in[3];
  const float* g1  = (const float*)d_in[4];
  const float* be1 = (const float*)d_in[5];
  const float* mu1 = (const float*)d_in[6];
  const float* vr1 = (const float*)d_in[7];
  const float* W2  = (const float*)d_in[8];
  const float* b2  = (const float*)d_in[9];
  const float* g2  = (const float*)d_in[10];
  const float* be2 = (const float*)d_in[11];
  const float* mu2 = (const float*)d_in[12];
  const float* vr2 = (const float*)d_in[13];
  const float* W3  = (const float*)d_in[14];
  const float* b3  = (const float*)d_in[15];
  const float* g3  = (const float*)d_in[16];
  const float* be3 = (const float*)d_in[17];
  const float* mu3 = (const float*)d_in[18];
  const float* vr3 = (const float*)d_in[19];

  float* out_newxyz = (float*)d_out;                       // (B,3,1024)
  float* out_points = (float*)d_out + (size_t)NB * 3 * NPT; // (B,128,1024)

  // workspace: idx (1 MB) | newxyz (96 KB) | coefA/B (2 KB)
  char*  ws        = (char*)d_ws;
  int*   idx_ws    = (int*)ws;
  float* newxyz_ws = (float*)(ws + (size_t)NB * NPT * NSAMP * sizeof(int));
  float* coefA     = newxyz_ws + (size_t)NB * NPT * 3;
  float* coefB     = coefA + 256;

  coef_kernel<<<1, 128, 0, stream>>>(b1, g1, be1, mu1, vr1, coefA, coefB, 64);
  coef_kernel<<<1, 128, 0, stream>>>(b2, g2, be2, mu2, vr2, coefA + 64, coefB + 64, 64);
  coef_kernel<<<1, 128, 0, stream>>>(b3, g3, be3, mu3, vr3, coefA + 128, coefB + 128, 128);

  size_t fps_smem = (size_t)(4 * NN + 64) * sizeof(float);   // 256 KB + scratch, in 320 KB LDS
  fps_kernel<<<NB, 1024, fps_smem, stream>>>(xyz, newxyz_ws, out_newxyz);

  ballquery_kernel<<<(NB * NPT) / 8, 256, 0, stream>>>(xyz, newxyz_ws, idx_ws);

  size_t mlp_smem = (size_t)(8 * 64 + 64 * 64 + 64 * 128 + 512 + 4 * (256 + 2 * 32 * HS))
                    * sizeof(float);                         // ~122 KB
  mlp_kernel<<<(NB * NPT) / 4, 128, mlp_smem, stream>>>(xyz, points, W1, W2, W3,
                                                        newxyz_ws, idx_ws, coefA, coefB,
                                                        out_points);
}